// NeighborIntegration_54305566491418
// MI455X (gfx1250) — compile-verified
//
#include <hip/hip_runtime.h>

// Problem constants from the reference: B=64, K=64, D=128
#define Bn 64
#define Kn 64
#define Dn 128

typedef float v2f __attribute__((ext_vector_type(2)));
typedef float v8f __attribute__((ext_vector_type(8)));

// ---------------------------------------------------------------------------
// Pass 1: For each (b,i): R[b,i,:] = 64*mask_c*concept[b,i,:]
//                                   + sum_j mask_n[j]*neighbor[b,i,j,:]
//         scale[b,i]     = 64*mask_c + sum_j mask_n[j]
// One block per (b,i). Tile (64x128 f32 = 32KB) staged in LDS so the 134MB
// neighbor tensor is read from HBM exactly once, fully coalesced (B128 loads).
// ---------------------------------------------------------------------------
__global__ void __launch_bounds__(256)
reduce_neighbors(const float* __restrict__ neighbor,
                 const float* __restrict__ concept_,
                 float* __restrict__ R,
                 float* __restrict__ scale) {
  __shared__ float tile[Kn * Dn];   // 32 KB
  __shared__ float rowsum[Kn];
  __shared__ float colbuf[256];
  __shared__ float csum;
  __shared__ float cntS;

  const int bi  = blockIdx.x;       // 0..4095  (= b*64 + i)
  const int tid = threadIdx.x;      // 0..255

  if (tid < Kn) rowsum[tid] = 0.f;
  if (tid == 0) { csum = 0.f; cntS = 0.f; }
  __syncthreads();

  // Stream the 64x128 tile to LDS; fold per-row sums (for the mask) into the
  // same pass via ds_add_f32 atomics.
  const float4* src =
      reinterpret_cast<const float4*>(neighbor) + (size_t)bi * (Kn * Dn / 4);
#pragma unroll
  for (int it = 0; it < 8; ++it) {
    const int g = tid + it * 256;                 // float4 index 0..2047
    float4 v = src[g];
    reinterpret_cast<float4*>(tile)[g] = v;
    atomicAdd(&rowsum[g >> 5], v.x + v.y + v.z + v.w);  // 32 float4 per row
  }
  if (tid < Dn / 4) {
    float4 c = reinterpret_cast<const float4*>(concept_)[(size_t)bi * (Dn / 4) + tid];
    atomicAdd(&csum, c.x + c.y + c.z + c.w);
  }
  __syncthreads();

  if (tid < Kn && rowsum[tid] != 0.f) atomicAdd(&cntS, 1.f);

  // Masked column sums: thread -> column d = tid&127, j-half = tid>>7.
  {
    const int d = tid & (Dn - 1);
    const int jbase = (tid >> 7) * 32;
    float acc = 0.f;
#pragma unroll
    for (int jj = 0; jj < 32; ++jj) {
      const int j = jbase + jj;
      acc += (rowsum[j] != 0.f) ? tile[j * Dn + d] : 0.f;
    }
    colbuf[tid] = acc;
  }
  __syncthreads();

  if (tid < Dn) {
    const float total = colbuf[tid] + colbuf[tid + Dn];
    const float maskc = (csum != 0.f) ? 1.f : 0.f;
    const float cv = concept_[(size_t)bi * Dn + tid];
    R[(size_t)bi * Dn + tid] = 64.f * maskc * cv + total;
    if (tid == 0) scale[bi] = 64.f * maskc + cntS;
  }
}

// ---------------------------------------------------------------------------
// Pass 2: Tstu[b, d] = stu[b,:] @ W1b[:, d] + b1[d]   (W1b = rows 128..255)
// Tiny (64x128 @ 128x128); one block per b, one thread per output column.
// stu[b*128+r] is wave-uniform per iteration -> scalar load.
// ---------------------------------------------------------------------------
__global__ void __launch_bounds__(128)
stu_gemm(const float* __restrict__ stu,
         const float* __restrict__ W1,
         const float* __restrict__ b1,
         float* __restrict__ Tstu) {
  const int b = blockIdx.x;
  const int d = threadIdx.x;
  float acc = b1[d];
#pragma unroll 4
  for (int r = 0; r < Dn; ++r)
    acc += stu[b * Dn + r] * W1[(Dn + r) * Dn + d];
  Tstu[b * Dn + d] = acc;
}

// ---------------------------------------------------------------------------
// Pass 3: out[m, n] = 0.5 * ( R[m,:] @ W1a[:, n] + scale[m]*Tstu[m>>6, n] )
// m = b*64+i in 0..4095, n in 0..127. One 16x16 output tile per wave,
// V_WMMA_F32_16X16X4_F32 over K=128 (32 wmma/tile). Accumulator is seeded
// with scale*Tstu so the epilogue is a pure 0.5 scale.
//
// Fragment layouts (ISA 7.12.2):
//   A 16x4 f32:  lanes 0-15 = M, v[0]=K0 v[1]=K1; lanes 16-31: v[0]=K2 v[1]=K3
//   B 4x16 f32:  lanes 0-15 = N, v[0]=K0 v[1]=K1; lanes 16-31: v[0]=K2 v[1]=K3
//   C/D 16x16:   VGPR v: M = v + 8*(lane>=16), N = lane&15
// ---------------------------------------------------------------------------
__global__ void __launch_bounds__(256)
wmma_gemm(const float* __restrict__ R,
          const float* __restrict__ W1,      // rows 0..127 = W1a
          const float* __restrict__ scale,
          const float* __restrict__ Tstu,
          float* __restrict__ out) {
  const int lane = threadIdx.x & 31;
  const int wave = threadIdx.x >> 5;   // 0..7 -> n-tile
  const int m0 = blockIdx.x * 16;      // 256 blocks -> 4096 rows
  const int n0 = wave * 16;

  const int half = lane >> 4;          // K-pair select
  const int l = lane & 15;
  const int col = n0 + l;              // N for B and C/D fragments

  v8f c;
#pragma unroll
  for (int v = 0; v < 8; ++v) {
    const int row = m0 + v + 8 * half;
    c[v] = scale[row] * Tstu[(row >> 6) * Dn + col];
  }

  const int arow = m0 + l;             // M for A fragment
#pragma unroll
  for (int k = 0; k < Dn; k += 4) {
    const int ka = k + 2 * half;       // this lane-half's K pair
    v2f a = *reinterpret_cast<const v2f*>(&R[arow * Dn + ka]);   // 8B aligned
    v2f b;
    b.x = W1[ka * Dn + col];
    b.y = W1[(ka + 1) * Dn + col];
    c = __builtin_amdgcn_wmma_f32_16x16x4_f32(
        /*neg_a=*/false, a, /*neg_b=*/false, b,
        /*c_mod=*/(short)0, c, /*reuse_a=*/false, /*reuse_b=*/false);
  }

#pragma unroll
  for (int v = 0; v < 8; ++v) {
    const int row = m0 + v + 8 * half;
    out[row * Dn + col] = 0.5f * c[v];
  }
}

// ---------------------------------------------------------------------------
// Inputs (setup_inputs order): stu_emb[64,128], concept_emb[64,64,128],
// neighbor_emb[64,64,64,128], W1[256,128], b1[128], Wn[256,1], bn[1].
// Wn/bn are provably dead (softmax over a size-1 axis == 1.0).
// Output: [64,64,128] f32.
// ---------------------------------------------------------------------------
extern "C" void kernel_launch(void* const* d_in, const int* in_sizes, int n_in,
                              void* d_out, int out_size, void* d_ws, size_t ws_size,
                              hipStream_t stream) {
  const float* stu      = (const float*)d_in[0];
  const float* concept_ = (const float*)d_in[1];
  const float* neighbor = (const float*)d_in[2];
  const float* W1       = (const float*)d_in[3];
  const float* b1       = (const float*)d_in[4];
  // d_in[5] (Wn), d_in[6] (bn): unused by the math.

  float* ws    = (float*)d_ws;
  float* R     = ws;                       // 4096*128 f32 = 2 MB
  float* scl   = R + 4096 * Dn;            // 4096 f32
  float* Tstu  = scl + 4096;               // 64*128 f32
  float* outp  = (float*)d_out;

  reduce_neighbors<<<dim3(Bn * Kn), dim3(256), 0, stream>>>(neighbor, concept_, R, scl);
  stu_gemm<<<dim3(Bn), dim3(Dn), 0, stream>>>(stu, W1, b1, Tstu);
  wmma_gemm<<<dim3(256), dim3(256), 0, stream>>>(R, W1, scl, Tstu, outp);
}